// Net_41420664602929
// MI455X (gfx1250) — compile-verified
//
#include <hip/hip_runtime.h>
#include <hip/hip_bf16.h>

typedef __attribute__((ext_vector_type(2))) float v2f;
typedef __attribute__((ext_vector_type(8))) float v8f;

// ---------------------------------------------------------------- utilities

__global__ void zero_f32(float* __restrict__ p, int n) {
    int i = blockIdx.x * blockDim.x + threadIdx.x;
    if (i < n) p[i] = 0.0f;
}

// deg_out[src[e]] += 1 ; deg_in[dst[e]] += 1
__global__ void degree_kernel(const int* __restrict__ src, const int* __restrict__ dst,
                              float* __restrict__ deg_out, float* __restrict__ deg_in, int E) {
    int e = blockIdx.x * blockDim.x + threadIdx.x;
    if (e >= E) return;
    unsafeAtomicAdd(&deg_out[src[e]], 1.0f);
    unsafeAtomicAdd(&deg_in[dst[e]], 1.0f);
}

// c = rsqrt(max(deg, 1))
__global__ void finish_c(float* __restrict__ c_out, float* __restrict__ c_in, int N) {
    int i = blockIdx.x * blockDim.x + threadIdx.x;
    if (i >= N) return;
    c_out[i] = rsqrtf(fmaxf(c_out[i], 1.0f));
    c_in[i]  = rsqrtf(fmaxf(c_in[i], 1.0f));
}

// agg[dst[e]][f..f+3] += x[src[e]][f..f+3] * c_out[src[e]]
__global__ void scatter_edges(const float* __restrict__ x, const float* __restrict__ c_out,
                              const int* __restrict__ src, const int* __restrict__ dst,
                              float* __restrict__ agg, int E, int F) {
    int t = blockIdx.x * blockDim.x + threadIdx.x;
    int groups = F >> 2;
    int e = t / groups;
    if (e >= E) return;
    int f = (t - e * groups) << 2;
    int s = src[e], d = dst[e];
    float cs = c_out[s];
    const float4 v = *(const float4*)(x + (size_t)s * F + f);
    float* ad = agg + (size_t)d * F + f;
    unsafeAtomicAdd(ad + 0, v.x * cs);
    unsafeAtomicAdd(ad + 1, v.y * cs);
    unsafeAtomicAdd(ad + 2, v.z * cs);
    unsafeAtomicAdd(ad + 3, v.w * cs);
}

// ------------------------------------------------ WMMA f32 GEMM
// One wave computes a 16 x 64 output slab: 4 adjacent 16x16 WMMA tiles sharing
// the same A fragment (A loaded once per k-step, reused 4x).
// out[m][n] = relu?( (A[m][:] * scale[m]) . W[:][n] + bias[n] )
// A: [M,K] row-major, W: [K,N] row-major.  M%16==0, N%64==0, K%4==0.
#define GEMM_NT 4
__global__ __launch_bounds__(32)
void gemm_wmma_f32(const float* __restrict__ A, const float* __restrict__ rowscale,
                   const float* __restrict__ W, const float* __restrict__ bias,
                   float* __restrict__ out, int M, int K, int N, int relu) {
    const int lane = threadIdx.x;      // 0..31, one full wave32
    const int half = lane >> 4;        // 0: K pair {k,k+1}; 1: K pair {k+2,k+3}
    const int l16  = lane & 15;
    const int ngroups = N >> 6;        // slabs of 64 output cols
    const int bid = blockIdx.x;
    const int tm  = (bid / ngroups) << 4;
    const int tn0 = (bid - (bid / ngroups) * ngroups) << 6;

    const int mA = tm + l16;           // A row for this lane (fixed across K loop)
    const float s = rowscale ? rowscale[mA] : 1.0f;

    const float* __restrict__ arow = A + (size_t)mA * K;
    v8f acc[GEMM_NT];
#pragma unroll
    for (int nt = 0; nt < GEMM_NT; ++nt) acc[nt] = (v8f){};

    for (int k = 0; k < K; k += 4) {
        const int ka = k + (half << 1);
        v2f a = *(const v2f*)(arow + ka);   // {A[m][ka], A[m][ka+1]}
        a.x *= s; a.y *= s;
        const float* __restrict__ w0 = W + (size_t)ka * N + tn0 + l16;
#pragma unroll
        for (int nt = 0; nt < GEMM_NT; ++nt) {
            v2f b;
            b.x = w0[nt * 16];          // W[ka  ][tn0 + 16*nt + l16]
            b.y = w0[nt * 16 + N];      // W[ka+1][tn0 + 16*nt + l16]
            acc[nt] = __builtin_amdgcn_wmma_f32_16x16x4_f32(
                          /*neg_a=*/false, a, /*neg_b=*/false, b,
                          /*c_mod=*/(short)0, acc[nt],
                          /*reuse_a=*/false, /*reuse_b=*/false);
        }
    }
    // D layout: VGPR r -> M = tm + r + 8*half, N = tile_n + l16
#pragma unroll
    for (int nt = 0; nt < GEMM_NT; ++nt) {
        const int n = tn0 + nt * 16 + l16;
        const float bv = bias[n];
#pragma unroll
        for (int r = 0; r < 8; ++r) {
            const int m = tm + r + (half << 3);
            float v = acc[nt][r] + bv;
            if (relu) v = fmaxf(v, 0.0f);
            out[(size_t)m * N + n] = v;
        }
    }
}

// ---------------------------------------------------------------- readout + MLP glue

__global__ void readout_sum(const float* __restrict__ y, const int* __restrict__ gid,
                            float* __restrict__ gsum, int Nn, int F) {
    int t = blockIdx.x * blockDim.x + threadIdx.x;
    int groups = F >> 2;
    int n = t / groups;
    if (n >= Nn) return;
    int f = (t - n * groups) << 2;
    int g = gid[n];
    const float4 v = *(const float4*)(y + (size_t)n * F + f);
    float* gr = gsum + (size_t)g * F + f;
    unsafeAtomicAdd(gr + 0, v.x);
    unsafeAtomicAdd(gr + 1, v.y);
    unsafeAtomicAdd(gr + 2, v.z);
    unsafeAtomicAdd(gr + 3, v.w);
}

__global__ void readout_cnt(const int* __restrict__ gid, float* __restrict__ gcnt, int Nn) {
    int n = blockIdx.x * blockDim.x + threadIdx.x;
    if (n >= Nn) return;
    unsafeAtomicAdd(&gcnt[gid[n]], 1.0f);
}

// mlp_in[g][c] = c<H ? mean ; c<H+3 ? feats_graph ; 0 (pad)
__global__ void build_mlp_in(const float* __restrict__ gsum, const float* __restrict__ gcnt,
                             const float* __restrict__ fg, float* __restrict__ out,
                             int G, int H, int KP) {
    int t = blockIdx.x * blockDim.x + threadIdx.x;
    if (t >= G * KP) return;
    int g = t / KP, c = t - g * KP;
    float v;
    if (c < H)            v = gsum[(size_t)g * H + c] / fmaxf(gcnt[g], 1.0f);
    else if (c < H + 3)   v = fg[g * 3 + (c - H)];
    else                  v = 0.0f;
    out[t] = v;
}

// pad Wl1 [Kin x N] -> [KP x N] with zero rows
__global__ void pad_w(const float* __restrict__ Wsrc, float* __restrict__ Wdst,
                      int Kin, int KP, int N) {
    int t = blockIdx.x * blockDim.x + threadIdx.x;
    if (t >= KP * N) return;
    int r = t / N, c = t - r * N;
    Wdst[t] = (r < Kin) ? Wsrc[(size_t)r * N + c] : 0.0f;
}

// out[g] = dot(t2[g], w) + b   (N=1 output layer)
__global__ void final_dot(const float* __restrict__ x, const float* __restrict__ w,
                          const float* __restrict__ b, float* __restrict__ out,
                          int G, int K) {
    int g = blockIdx.x * blockDim.x + threadIdx.x;
    if (g >= G) return;
    const float* r = x + (size_t)g * K;
    float acc = 0.0f;
    for (int k = 0; k < K; ++k) acc = fmaf(r[k], w[k], acc);
    out[g] = acc + b[0];
}

// ---------------------------------------------------------------- launcher

extern "C" void kernel_launch(void* const* d_in, const int* in_sizes, int n_in,
                              void* d_out, int out_size, void* d_ws, size_t ws_size,
                              hipStream_t stream) {
    const float* feats_node  = (const float*)d_in[0];
    const float* feats_graph = (const float*)d_in[1];
    const float* W1 = (const float*)d_in[2];  const float* b1 = (const float*)d_in[3];
    const float* W2 = (const float*)d_in[4];  const float* b2 = (const float*)d_in[5];
    const float* W3 = (const float*)d_in[6];  const float* b3 = (const float*)d_in[7];
    const float* Wl1 = (const float*)d_in[8]; const float* bl1 = (const float*)d_in[9];
    const float* Wl2 = (const float*)d_in[10]; const float* bl2 = (const float*)d_in[11];
    const float* Wl3 = (const float*)d_in[12]; const float* bl3 = (const float*)d_in[13];
    const int* src = (const int*)d_in[14];
    const int* dst = (const int*)d_in[15];
    const int* gid = (const int*)d_in[16];

    const int N  = in_sizes[16];          // 50000 nodes (multiple of 16)
    const int E  = in_sizes[14];          // 600000 edges
    const int G  = in_sizes[1] / 3;       // 2000 graphs (multiple of 16)
    const int F0 = in_sizes[0] / N;       // 64 input feats
    const int H  = in_sizes[3];           // 128 hidden
    const int H2 = 2 * H;                 // 256
    const int KP = H + 4;                 // 132 = padded (H+3), multiple of 4

    // workspace carve-up (floats)
    float* ws = (float*)d_ws;
    float* c_out  = ws;  ws += N;
    float* c_in   = ws;  ws += N;               // contiguous with c_out
    float* agg    = ws;  ws += (size_t)N * H;   // scatter target
    float* xbuf   = ws;  ws += (size_t)N * H;   // layer output (ping-pong vs agg)
    float* gsum   = ws;  ws += (size_t)G * H;
    float* gcnt   = ws;  ws += G;               // contiguous with gsum
    float* mlp_in = ws;  ws += (size_t)G * KP;
    float* wl1p   = ws;  ws += (size_t)KP * H2;
    float* t1     = ws;  ws += (size_t)G * H2;
    float* t2     = ws;  ws += (size_t)G * H2;

    const int B = 256;
    auto cdiv = [](int a, int b) { return (a + b - 1) / b; };

    // --- degree normalization coefficients
    zero_f32<<<cdiv(2 * N, B), B, 0, stream>>>(c_out, 2 * N);
    degree_kernel<<<cdiv(E, B), B, 0, stream>>>(src, dst, c_out, c_in, E);
    finish_c<<<cdiv(N, B), B, 0, stream>>>(c_out, c_in, N);

    // --- gconv layer 1: feats_node (F0) -> xbuf (H), relu
    zero_f32<<<cdiv(N * F0, B), B, 0, stream>>>(agg, N * F0);
    scatter_edges<<<cdiv(E * (F0 >> 2), B), B, 0, stream>>>(feats_node, c_out, src, dst, agg, E, F0);
    gemm_wmma_f32<<<(N / 16) * (H / 64), 32, 0, stream>>>(agg, c_in, W1, b1, xbuf, N, F0, H, 1);

    // --- gconv layer 2: xbuf -> xbuf, relu
    zero_f32<<<cdiv(N * H, B), B, 0, stream>>>(agg, N * H);
    scatter_edges<<<cdiv(E * (H >> 2), B), B, 0, stream>>>(xbuf, c_out, src, dst, agg, E, H);
    gemm_wmma_f32<<<(N / 16) * (H / 64), 32, 0, stream>>>(agg, c_in, W2, b2, xbuf, N, H, H, 1);

    // --- gconv layer 3: xbuf -> xbuf, no relu
    zero_f32<<<cdiv(N * H, B), B, 0, stream>>>(agg, N * H);
    scatter_edges<<<cdiv(E * (H >> 2), B), B, 0, stream>>>(xbuf, c_out, src, dst, agg, E, H);
    gemm_wmma_f32<<<(N / 16) * (H / 64), 32, 0, stream>>>(agg, c_in, W3, b3, xbuf, N, H, H, 0);

    // --- mean readout over graphs
    zero_f32<<<cdiv(G * H + G, B), B, 0, stream>>>(gsum, G * H + G);   // gsum + gcnt
    readout_sum<<<cdiv(N * (H >> 2), B), B, 0, stream>>>(xbuf, gid, gsum, N, H);
    readout_cnt<<<cdiv(N, B), B, 0, stream>>>(gid, gcnt, N);
    build_mlp_in<<<cdiv(G * KP, B), B, 0, stream>>>(gsum, gcnt, feats_graph, mlp_in, G, H, KP);

    // --- MLP head
    pad_w<<<cdiv(KP * H2, B), B, 0, stream>>>(Wl1, wl1p, H + 3, KP, H2);
    gemm_wmma_f32<<<(G / 16) * (H2 / 64), 32, 0, stream>>>(mlp_in, nullptr, wl1p, bl1, t1, G, KP, H2, 1);
    gemm_wmma_f32<<<(G / 16) * (H2 / 64), 32, 0, stream>>>(t1, nullptr, Wl2, bl2, t2, G, H2, H2, 1);
    final_dot<<<cdiv(G, B), B, 0, stream>>>(t2, Wl3, bl3, (float*)d_out, G, H2);
}